// SpatialAttention_37520834298604
// MI455X (gfx1250) — compile-verified
//
#include <hip/hip_runtime.h>
#include <hip/hip_bf16.h>
#include <math.h>

typedef __attribute__((ext_vector_type(2))) float v2f;
typedef __attribute__((ext_vector_type(4))) float v4f;
typedef __attribute__((ext_vector_type(8))) float v8f;

#define BATCH 8
#define CHAN  128
#define HDIM  256
#define WDIM  256
#define HW    (HDIM * WDIM)          // 65536
#define KSZ   7

// ---------------------------------------------------------------------------
// Kernel 1: channel max-pool.  [B,C,H,W] -> [B,H,W].  Pure streaming read of
// 256MB with NT hint (don't thrash L2), tiny 2MB write.
// Each thread handles 4 consecutive pixels (b128 loads), loops over 128 chans.
// ---------------------------------------------------------------------------
__global__ void __launch_bounds__(256) chanmax_kernel(const float* __restrict__ in,
                                                      float* __restrict__ pooled) {
    const int gid = blockIdx.x * 256 + threadIdx.x;        // 131072 threads
    const int b   = gid >> 14;                             // 16384 quads per batch
    const int p4  = gid & 16383;                           // quad index in plane
    const size_t base = (size_t)b * CHAN * HW + (size_t)p4 * 4;

    const v4f* in4 = (const v4f*)(in + base);
    v4f m = __builtin_nontemporal_load(in4);               // c = 0
    #pragma unroll 4
    for (int c = 1; c < CHAN; ++c) {
        v4f v = __builtin_nontemporal_load(in4 + (size_t)c * (HW / 4));
        m.x = fmaxf(m.x, v.x);
        m.y = fmaxf(m.y, v.y);
        m.z = fmaxf(m.z, v.z);
        m.w = fmaxf(m.w, v.w);
    }
    *(v4f*)(pooled + (size_t)b * HW + (size_t)p4 * 4) = m;
}

// ---------------------------------------------------------------------------
// Kernel 2: 7x7 conv (+bias, sigmoid) on pooled map via V_WMMA_F32_16X16X4_F32.
//
// For a 16x16 output tile at (R, Cc):
//   out[y,x] = sum_dy sum_c  P[R-3 + y+dy, Cc-3 + c] * T_dy[c, x]
// where T_dy[c,x] = w[dy, c-x] for 0 <= c-x <= 6, else 0   (c in 0..21, pad 24)
// -> 7 dy * 6 K4-chunks = 42 WMMAs, all accumulation in f32.
//
// Block = 256 threads = 8 waves; block covers 16 rows x 128 cols (8 tiles).
// LDS: pooled halo patch 22 x 136 (+Toeplitz table 7x16x24, transposed so a
// lane's B fragment is one aligned ds_load_b64).
// ---------------------------------------------------------------------------
#define PSTRIDE 136   // 134 used cols (128 + 6 halo) padded to even-x8B stride
#define PROWS   22

__global__ void __launch_bounds__(256) conv_sigmoid_wmma_kernel(
        const float* __restrict__ pooled,
        const float* __restrict__ cw,     // 49 weights, row-major [dy][dx]
        const float* __restrict__ cb,     // 1 bias
        float* __restrict__ attn) {
    __shared__ float patch[PROWS * PSTRIDE];   // 11968 B
    __shared__ float btr[7 * 16 * 24];         // Toeplitz, [dy][x][c], 10752 B

    const int tid = threadIdx.x;
    const int bid = blockIdx.x;                // 256 blocks
    const int b       = bid >> 5;              // 8 batches
    const int rem     = bid & 31;
    const int tileRow = (rem >> 1) * 16;       // 16 row-bands
    const int colBase = (rem & 1) * 128;       // 2 col-bands of 128

    // --- cooperative fill: pooled halo patch (zero outside image / pad cols)
    const float* pb = pooled + (size_t)b * HW;
    const int rowBase = tileRow - 3;
    const int colBase3 = colBase - 3;
    for (int idx = tid; idx < PROWS * PSTRIDE; idx += 256) {
        const int r = idx / PSTRIDE;
        const int c = idx - r * PSTRIDE;
        const int gr = rowBase + r;
        const int gc = colBase3 + c;
        float v = 0.0f;
        if ((unsigned)gr < HDIM && (unsigned)gc < WDIM && c < 134)
            v = pb[gr * WDIM + gc];
        patch[idx] = v;
    }
    // --- cooperative fill: transposed Toeplitz weight table btr[dy][x][c]
    for (int idx = tid; idx < 7 * 16 * 24; idx += 256) {
        const int dy = idx / (16 * 24);
        const int r2 = idx - dy * (16 * 24);
        const int x  = r2 / 24;
        const int c  = r2 - x * 24;
        const int dx = c - x;
        float v = 0.0f;
        if (dx >= 0 && dx <= 6 && c < 22) v = cw[dy * 7 + dx];
        btr[idx] = v;
    }
    __syncthreads();

    // --- per-wave 16x16 tile via 42 x V_WMMA_F32_16X16X4_F32
    const int wave = tid >> 5;
    const int lane = tid & 31;
    const int half = lane >> 4;     // 0: K0/K1 pair, 1: K2/K3 pair
    const int n16  = lane & 15;     // A: row m; B: col n; D: col n
    const int waveCol = wave * 16;  // tile col offset inside patch

    v8f acc = {};
    #pragma unroll
    for (int dy = 0; dy < 7; ++dy) {
        const float* arow = &patch[(n16 + dy) * PSTRIDE + waveCol + 2 * half];
        const float* brow = &btr[(dy * 16 + n16) * 24 + 2 * half];
        #pragma unroll
        for (int j = 0; j < 6; ++j) {
            v2f a  = *(const v2f*)(arow + 4 * j);   // A frag: {K0,K1}/{K2,K3}
            v2f bm = *(const v2f*)(brow + 4 * j);   // B frag: {K0,K1}/{K2,K3}
            acc = __builtin_amdgcn_wmma_f32_16x16x4_f32(
                false, a, false, bm, (short)0, acc, false, false);
        }
    }

    // --- bias + sigmoid, scatter D (VGPR g -> rows g / g+8, col = n16)
    const float bias = cb[0];
    float* ab = attn + (size_t)b * HW;
    #pragma unroll
    for (int g = 0; g < 8; ++g) {
        const int row = tileRow + g + 8 * half;
        const int col = colBase + waveCol + n16;
        const float v = acc[g] + bias;
        ab[row * WDIM + col] = 1.0f / (1.0f + expf(-v));
    }
}

// ---------------------------------------------------------------------------
// Kernel 3: out = in * attn (broadcast over channels).  Pure streaming:
// NT for the 256MB in/out streams; attn (2MB) stays cached in L2.
// ---------------------------------------------------------------------------
__global__ void __launch_bounds__(256) bcast_mul_kernel(const float* __restrict__ in,
                                                        const float* __restrict__ attn,
                                                        float* __restrict__ out) {
    const size_t gid = (size_t)blockIdx.x * 256 + threadIdx.x;  // 16,777,216
    const size_t e   = gid * 4;                                 // element offset
    const size_t bc  = e >> 16;                                 // plane index
    const size_t hw  = e & 65535;
    const size_t bb  = bc >> 7;                                 // batch
    const size_t ai  = (bb << 16) + hw;

    v4f x = __builtin_nontemporal_load((const v4f*)(in + e));
    v4f a = *(const v4f*)(attn + ai);
    v4f r;
    r.x = x.x * a.x; r.y = x.y * a.y; r.z = x.z * a.z; r.w = x.w * a.w;
    __builtin_nontemporal_store(r, (v4f*)(out + e));
}

// ---------------------------------------------------------------------------
extern "C" void kernel_launch(void* const* d_in, const int* in_sizes, int n_in,
                              void* d_out, int out_size, void* d_ws, size_t ws_size,
                              hipStream_t stream) {
    const float* x  = (const float*)d_in[0];   // [8,128,256,256]
    const float* cw = (const float*)d_in[1];   // [1,1,7,7]
    const float* cb = (const float*)d_in[2];   // [1]
    float* out = (float*)d_out;

    float* pooled = (float*)d_ws;              // 8*65536 f32 = 2MB
    float* attn   = pooled + BATCH * HW;       // 2MB more

    // 1) channel max: 131072 threads
    chanmax_kernel<<<512, 256, 0, stream>>>(x, pooled);
    // 2) conv + sigmoid via WMMA: 256 blocks x 8 tiles
    conv_sigmoid_wmma_kernel<<<256, 256, 0, stream>>>(pooled, cw, cb, attn);
    // 3) broadcast multiply: 67,108,864 elems / 4 per thread
    bcast_mul_kernel<<<65536, 256, 0, stream>>>(x, attn, out);
}